// AttractorCore_14010183320239
// MI455X (gfx1250) — compile-verified
//
#include <hip/hip_runtime.h>
#include <math.h>

typedef __attribute__((ext_vector_type(2))) float v2f;
typedef __attribute__((ext_vector_type(8))) float v8f;

#define BB 4
#define NN 1024
#define DD 128
#define KK 10
#define MM 4136          // B*N + B*K
#define MP 4160          // padded to multiple of 64
#define ITERS 50
#define JTILES (MP / 16) // 260
#define JSPLIT 8
#define JPER 33          // ceil(260/8)
#define GCONST 1e-3f
#define DTC 0.1f
#define DAMPC 0.95f
#define EPSC 1e-6f

// ---- workspace layout (float offsets) ----
#define OFF_RA   0
#define OFF_VA   (MP * DD)
#define OFF_RB   (2 * MP * DD)
#define OFF_VB   (3 * MP * DD)
#define OFF_AACC (4 * MP * DD)
#define OFF_MASS (5 * MP * DD)
#define OFF_SQ   (5 * MP * DD + MP)
#define OFF_WSUM (5 * MP * DD + 2 * MP)
#define OFF_FEAT (5 * MP * DD + 3 * MP)   // 4 * 388
#define OFF_PD   (OFF_FEAT + 4 * 388)     // 8 slots

// ------------------------------------------------------------------
// init: build padded state, masses, reset pdist accumulators
// ------------------------------------------------------------------
__global__ __launch_bounds__(128)
void init_kernel(const float* __restrict__ x, const float* __restrict__ ap,
                 const float* __restrict__ am, const float* __restrict__ iv,
                 float* __restrict__ rA, float* __restrict__ vA,
                 float* __restrict__ mass, float* __restrict__ pd) {
  int row = blockIdx.x, d = threadIdx.x;
  float rv = 0.0f, vv = 0.0f, m = 0.0f;
  if (row < BB * NN) {
    rv = x[row * DD + d];
    m = 0.1f;
  } else if (row < MM) {
    int k = (row - BB * NN) % KK;
    rv = ap[k * DD + d];
    vv = iv[k * DD + d];
    m = am[k];
  }
  rA[row * DD + d] = rv;
  vA[row * DD + d] = vv;
  if (d == 0) mass[row] = m;
  if (row == 0 && d == 0) {
    pd[0] = 0.0f;                  // sum of distances
    pd[1] = 0.0f;                  // sum of squared distances
    ((int*)pd)[2] = 0x7f800000;    // min (bits of +inf; nonneg float order == int order)
    ((int*)pd)[3] = 0;             // max (bits of 0.0f)
  }
}

// ------------------------------------------------------------------
// prep (per iteration): sq[row] = |r_row|^2, zero a_acc and wsum
// ------------------------------------------------------------------
__global__ __launch_bounds__(128)
void prep_kernel(const float* __restrict__ rA, float* __restrict__ sq,
                 float* __restrict__ aacc, float* __restrict__ wsum) {
  __shared__ float red[128];
  int row = blockIdx.x, t = threadIdx.x;
  float v = rA[row * DD + t];
  red[t] = v * v;
  aacc[row * DD + t] = 0.0f;
  __syncthreads();
  for (int st = 64; st > 0; st >>= 1) {
    if (t < st) red[t] += red[t + st];
    __syncthreads();
  }
  if (t == 0) {
    sq[row] = red[0];
    wsum[row] = 0.0f;
  }
}

// ------------------------------------------------------------------
// gravity: all-pairs via fp32 WMMA (16x16x4).
//   S = Ri * Rj^T  (Gram -> squared distances)
//   Acc += w * Rj  (weighted position sum)
// Block = 4 waves, each wave owns one 16-row i-tile; gridDim.y splits j.
// ------------------------------------------------------------------
__global__ __launch_bounds__(128)
void grav_kernel(const float* __restrict__ r, const float* __restrict__ sq,
                 const float* __restrict__ mass, float* __restrict__ aacc,
                 float* __restrict__ wsum) {
  __shared__ float lrj[16 * 132];    // j-tile, row-major, padded stride
  __shared__ float lrjT[128 * 20];   // j-tile, col-major, padded stride
  __shared__ float lw[4][16 * 18];   // per-wave 16x16 weight tile

  const int tid  = threadIdx.x;
  const int wv   = tid >> 5;
  const int lane = tid & 31;
  const int half = lane >> 4;        // 0: K/M low half, 1: high half
  const int l15  = lane & 15;
  const int i0   = (blockIdx.x * 4 + wv) * 16;
  const int jt0  = blockIdx.y * JPER;
  int jt1 = jt0 + JPER;
  if (jt1 > JTILES) jt1 = JTILES;

  // Preload A fragments for the i-tile: 32 chunks of 16x4 f32.
  // A layout (16x4 f32): lane<16 holds row l15, K = kc*4 + {0,1};
  //                      lane>=16 holds row l15, K = kc*4 + {2,3}.
  v2f afrag[32];
#pragma unroll
  for (int kc = 0; kc < 32; ++kc) {
    const float* p = &r[(i0 + l15) * DD + kc * 4 + 2 * half];
    afrag[kc] = *(const v2f*)p;
  }
  float sqi[8];
#pragma unroll
  for (int p = 0; p < 8; ++p) sqi[p] = sq[i0 + p + 8 * half];

  v8f acc[8];
#pragma unroll
  for (int t = 0; t < 8; ++t) acc[t] = (v8f){0, 0, 0, 0, 0, 0, 0, 0};
  float wsp[8] = {0, 0, 0, 0, 0, 0, 0, 0};

  for (int jt = jt0; jt < jt1; ++jt) {
    const int j0 = jt * 16;
    __syncthreads();
    // Cooperative load of the 16x128 j-tile, both layouts.
#pragma unroll
    for (int q = 0; q < 4; ++q) {
      int e = (tid + q * 128) * 4;
      int rr = e >> 7, cc = e & 127;
      const float* src = &r[(j0 + rr) * DD + cc];
      float x0 = src[0], x1 = src[1], x2 = src[2], x3 = src[3];
      float* drow = &lrj[rr * 132 + cc];
      drow[0] = x0; drow[1] = x1; drow[2] = x2; drow[3] = x3;
      lrjT[(cc + 0) * 20 + rr] = x0;
      lrjT[(cc + 1) * 20 + rr] = x1;
      lrjT[(cc + 2) * 20 + rr] = x2;
      lrjT[(cc + 3) * 20 + rr] = x3;
    }
    __syncthreads();

    // ---- Gram: S(16x16) = Ri * Rj^T over K=128 in chunks of 4 ----
    // B frag (4x16): lane holds column l15, rows {0,1}+2*half of the chunk,
    // i.e. Rj[l15][k + 2*half .. +1]  -> contiguous b64 from row-major tile.
    v8f S = (v8f){0, 0, 0, 0, 0, 0, 0, 0};
#pragma unroll
    for (int kc = 0; kc < 32; ++kc) {
      v2f bfrag = *(const v2f*)&lrj[l15 * 132 + kc * 4 + 2 * half];
      S = __builtin_amdgcn_wmma_f32_16x16x4_f32(false, afrag[kc], false, bfrag,
                                                (short)0, S, false, false);
    }

    // ---- weights: w = m_j * d2^-1.5, diag masked ----
    const int jg = j0 + l15;
    const float sqj = sq[jg];
    const float mj = mass[jg];
#pragma unroll
    for (int p = 0; p < 8; ++p) {
      const int ig = i0 + p + 8 * half;
      float d2 = sqi[p] + sqj - 2.0f * S[p];
      d2 = fmaxf(d2, EPSC);
      float inv3 = 1.0f / (d2 * sqrtf(d2));
      float w = (ig == jg) ? 0.0f : inv3 * mj;
      wsp[p] += w;
      lw[wv][(p + 8 * half) * 18 + l15] = w;  // C-layout -> LDS
    }
    __syncthreads();

    // ---- Acc(16x128) += w(16x16) * Rj(16x128), 8 N-tiles x 4 K-chunks ----
#pragma unroll
    for (int t = 0; t < 8; ++t) {
#pragma unroll
      for (int kk = 0; kk < 4; ++kk) {
        v2f aw = *(const v2f*)&lw[wv][l15 * 18 + kk * 4 + 2 * half];
        v2f bf = *(const v2f*)&lrjT[(t * 16 + l15) * 20 + kk * 4 + 2 * half];
        acc[t] = __builtin_amdgcn_wmma_f32_16x16x4_f32(false, aw, false, bf,
                                                       (short)0, acc[t], false, false);
      }
    }
  }

  // Row sums of w: reduce across the 16 lanes of each half.
#pragma unroll
  for (int p = 0; p < 8; ++p) {
    float v = wsp[p];
    v += __shfl_xor(v, 1, 32);
    v += __shfl_xor(v, 2, 32);
    v += __shfl_xor(v, 4, 32);
    v += __shfl_xor(v, 8, 32);
    wsp[p] = v;
  }
  if (l15 == 0) {
#pragma unroll
    for (int p = 0; p < 8; ++p)
      atomicAdd(&wsum[i0 + p + 8 * half], wsp[p]);
  }
#pragma unroll
  for (int t = 0; t < 8; ++t)
#pragma unroll
    for (int p = 0; p < 8; ++p)
      atomicAdd(&aacc[(i0 + p + 8 * half) * DD + t * 16 + l15], acc[t][p]);
}

// ------------------------------------------------------------------
// update: a = G*(Acc - wsum*r); v += a*dt; r += v*dt   (A -> B buffers)
// ------------------------------------------------------------------
__global__ __launch_bounds__(256)
void update_kernel(const float* __restrict__ rA, const float* __restrict__ vA,
                   const float* __restrict__ aacc, const float* __restrict__ wsum,
                   float* __restrict__ rB, float* __restrict__ vB) {
  int idx = blockIdx.x * blockDim.x + threadIdx.x;
  if (idx >= MM * DD) return;
  int row = idx >> 7;
  float a = GCONST * (aacc[idx] - wsum[row] * rA[idx]);
  float v = vA[idx] + a * DTC;
  float rr = rA[idx] + v * DTC;
  vB[idx] = v;
  rB[idx] = rr;
}

// ------------------------------------------------------------------
// pointwise: boundary MLP, optional Hamiltonian grads, damping, clip
// one wave per particle (B -> A buffers)
// ------------------------------------------------------------------
__global__ __launch_bounds__(32)
void point_kernel(const float* __restrict__ rB, const float* __restrict__ vB,
                  float* __restrict__ rA, float* __restrict__ vA,
                  const float* __restrict__ bw1, const float* __restrict__ bb1,
                  const float* __restrict__ bw2, const float* __restrict__ bb2,
                  const float* __restrict__ bw3, const float* __restrict__ bb3,
                  const float* __restrict__ pw1, const float* __restrict__ pb1,
                  const float* __restrict__ pw2,
                  const float* __restrict__ kw1, const float* __restrict__ kb1,
                  const float* __restrict__ kw2, int do_ham) {
  const int row = blockIdx.x;
  const int lane = threadIdx.x;
  __shared__ float cr[128], cv[128], h1[64], h2[64], tb[256];
#pragma unroll
  for (int q = 0; q < 4; ++q) {
    cr[lane + 32 * q] = rB[row * DD + lane + 32 * q];
    cv[lane + 32 * q] = vB[row * DD + lane + 32 * q];
  }
  __syncthreads();
  // 256 -> 64, tanh
#pragma unroll
  for (int uu = 0; uu < 2; ++uu) {
    int u = lane + 32 * uu;
    float a = bb1[u];
    for (int c = 0; c < 256; ++c) {
      float x = (c < 128) ? cr[c] : cv[c - 128];
      a += x * bw1[c * 64 + u];
    }
    h1[u] = tanhf(a);
  }
  __syncthreads();
  // 64 -> 64, tanh
#pragma unroll
  for (int uu = 0; uu < 2; ++uu) {
    int u = lane + 32 * uu;
    float a = bb2[u];
    for (int c = 0; c < 64; ++c) a += h1[c] * bw2[c * 64 + u];
    h2[u] = tanhf(a);
  }
  __syncthreads();
  // 64 -> 128; r += 0.1*tanh(.)
  float rr[4], vv[4];
#pragma unroll
  for (int q = 0; q < 4; ++q) {
    int d = lane + 32 * q;
    float a = bb3[d];
    for (int c = 0; c < 64; ++c) a += h2[c] * bw3[c * DD + d];
    rr[q] = cr[d] + 0.1f * tanhf(a);
    vv[q] = cv[d];
  }
  __syncthreads();
#pragma unroll
  for (int q = 0; q < 4; ++q) cr[lane + 32 * q] = rr[q];
  __syncthreads();
  if (do_ham) {
    // grad of potential wrt r:  g_d = sum_j (1-h_j^2)*pw2_j*pw1[d][j]
#pragma unroll
    for (int jj = 0; jj < 8; ++jj) {
      int j = lane + 32 * jj;
      float a = pb1[j];
      for (int c = 0; c < 128; ++c) a += cr[c] * pw1[c * 256 + j];
      float h = tanhf(a);
      tb[j] = (1.0f - h * h) * pw2[j];
    }
    __syncthreads();
#pragma unroll
    for (int q = 0; q < 4; ++q) {
      int d = lane + 32 * q;
      float g = 0.0f;
      for (int j = 0; j < 256; ++j) g += tb[j] * pw1[d * 256 + j];
      rr[q] -= 0.01f * g;  // r += 0.01*dp, dp = -grad
    }
    __syncthreads();
    // grad of kinetic wrt v
#pragma unroll
    for (int jj = 0; jj < 8; ++jj) {
      int j = lane + 32 * jj;
      float a = kb1[j];
      for (int c = 0; c < 128; ++c) a += cv[c] * kw1[c * 256 + j];
      float h = tanhf(a);
      tb[j] = (1.0f - h * h) * kw2[j];
    }
    __syncthreads();
#pragma unroll
    for (int q = 0; q < 4; ++q) {
      int d = lane + 32 * q;
      float g = 0.0f;
      for (int j = 0; j < 256; ++j) g += tb[j] * kw1[d * 256 + j];
      vv[q] += 0.01f * g;
    }
  }
  // damping + norm clip
  float ss = 0.0f;
#pragma unroll
  for (int q = 0; q < 4; ++q) {
    vv[q] *= DAMPC;
    ss += rr[q] * rr[q];
  }
  ss += __shfl_xor(ss, 16, 32);
  ss += __shfl_xor(ss, 8, 32);
  ss += __shfl_xor(ss, 4, 32);
  ss += __shfl_xor(ss, 2, 32);
  ss += __shfl_xor(ss, 1, 32);
  float nrm = sqrtf(ss);
  float scale = fminf(1.0f, 10.0f / fmaxf(nrm, 1e-12f));
#pragma unroll
  for (int q = 0; q < 4; ++q) {
    int d = lane + 32 * q;
    rA[row * DD + d] = rr[q] * scale;
    vA[row * DD + d] = vv[q];
  }
}

// ------------------------------------------------------------------
// stats: com / var / skew per batch (mass=1 for particles, am for attractors)
// ------------------------------------------------------------------
__global__ __launch_bounds__(128)
void stats1_kernel(const float* __restrict__ r, const float* __restrict__ am,
                   float* __restrict__ feat) {
  int b = blockIdx.x, d = threadIdx.x;
  float tot = (float)NN;
  for (int k = 0; k < KK; ++k) tot += am[k];
  float s = 0.0f;
  for (int n = 0; n < NN; ++n) s += r[(b * NN + n) * DD + d];
  for (int k = 0; k < KK; ++k) s += am[k] * r[(BB * NN + b * KK + k) * DD + d];
  float com = s / tot;
  float sv = 0.0f, sk = 0.0f;
  for (int n = 0; n < NN; ++n) {
    float c = r[(b * NN + n) * DD + d] - com;
    sv += c * c;
    sk += c * c * c;
  }
  for (int k = 0; k < KK; ++k) {
    float c = r[(BB * NN + b * KK + k) * DD + d] - com;
    sv += c * c;
    sk += c * c * c;
  }
  feat[b * 388 + d] = com;
  feat[b * 388 + 128 + d] = sv / (float)(NN + KK);
  feat[b * 388 + 256 + d] = sk / (float)(NN + KK);
}

__device__ __forceinline__ int row_of_p0(int n) {
  return (n < NN) ? n : (BB * NN + (n - NN));
}

// pdist over batch 0: accumulate sum, sumsq, min, max over all pairs
__global__ __launch_bounds__(128)
void pdist_kernel(const float* __restrict__ r, float* __restrict__ pd) {
  const int i = blockIdx.x;
  const int t = threadIdx.x;
  __shared__ float pi[128];
  __shared__ float red[128];
  pi[t] = r[row_of_p0(i) * DD + t];
  __syncthreads();
  float s = 0.0f, q2 = 0.0f, mn = 3.4e38f, mx = 0.0f;
  for (int j = i + 1 + t; j < NN + KK; j += 128) {
    const float* pj = &r[row_of_p0(j) * DD];
    float d2 = 0.0f;
    for (int c = 0; c < DD; ++c) {
      float df = pi[c] - pj[c];
      d2 += df * df;
    }
    float dd = sqrtf(fmaxf(d2, 0.0f));
    s += dd; q2 += dd * dd;
    mn = fminf(mn, dd); mx = fmaxf(mx, dd);
  }
  red[t] = s; __syncthreads();
  for (int st = 64; st > 0; st >>= 1) { if (t < st) red[t] += red[t + st]; __syncthreads(); }
  float bs = red[0]; __syncthreads();
  red[t] = q2; __syncthreads();
  for (int st = 64; st > 0; st >>= 1) { if (t < st) red[t] += red[t + st]; __syncthreads(); }
  float bq = red[0]; __syncthreads();
  red[t] = mn; __syncthreads();
  for (int st = 64; st > 0; st >>= 1) { if (t < st) red[t] = fminf(red[t], red[t + st]); __syncthreads(); }
  float bmn = red[0]; __syncthreads();
  red[t] = mx; __syncthreads();
  for (int st = 64; st > 0; st >>= 1) { if (t < st) red[t] = fmaxf(red[t], red[t + st]); __syncthreads(); }
  float bmx = red[0];
  if (t == 0 && i < NN + KK - 1) {
    atomicAdd(&pd[0], bs);
    atomicAdd(&pd[1], bq);
    atomicMin((int*)&pd[2], __float_as_int(bmn));
    atomicMax((int*)&pd[3], __float_as_int(bmx));
  }
}

// finalize stats + output projection: out[b] = feat[b] @ fw + fb
__global__ __launch_bounds__(512)
void final_kernel(const float* __restrict__ pd, float* __restrict__ feat,
                  const float* __restrict__ fw, const float* __restrict__ fb,
                  float* __restrict__ out) {
  __shared__ float s4[4];
  int t = threadIdx.x;
  if (t == 0) {
    const float P = 534061.0f;  // 1034*1033/2
    float sum = pd[0], q2 = pd[1];
    float mean = sum / P;
    float var = (q2 - sum * sum / P) / (P - 1.0f);
    s4[0] = mean;
    s4[1] = sqrtf(fmaxf(var, 0.0f));
    s4[2] = pd[2];  // bits of nonneg float == float value
    s4[3] = pd[3];
  }
  __syncthreads();
  if (t < 16) {
    int b = t >> 2, q = t & 3;
    feat[b * 388 + 384 + q] = s4[q];
  }
  __syncthreads();
  int b = t >> 7, d = t & 127;
  float a = fb[d];
  for (int c = 0; c < 388; ++c) a += feat[b * 388 + c] * fw[c * DD + d];
  out[b * DD + d] = a;
}

// ------------------------------------------------------------------
extern "C" void kernel_launch(void* const* d_in, const int* in_sizes, int n_in,
                              void* d_out, int out_size, void* d_ws, size_t ws_size,
                              hipStream_t stream) {
  (void)in_sizes; (void)n_in; (void)out_size; (void)ws_size;
  const float* x   = (const float*)d_in[0];
  const float* ap  = (const float*)d_in[1];
  const float* am  = (const float*)d_in[2];
  const float* iv  = (const float*)d_in[3];
  const float* pw1 = (const float*)d_in[4];
  const float* pb1 = (const float*)d_in[5];
  const float* pw2 = (const float*)d_in[6];
  // pb2 (d_in[7]) shifts the scalar potential only; gradient unaffected
  const float* kw1 = (const float*)d_in[8];
  const float* kb1 = (const float*)d_in[9];
  const float* kw2 = (const float*)d_in[10];
  // kb2 (d_in[11]) likewise unused
  const float* bw1 = (const float*)d_in[12];
  const float* bb1 = (const float*)d_in[13];
  const float* bw2 = (const float*)d_in[14];
  const float* bb2 = (const float*)d_in[15];
  const float* bw3 = (const float*)d_in[16];
  const float* bb3 = (const float*)d_in[17];
  const float* fw  = (const float*)d_in[18];
  const float* fb  = (const float*)d_in[19];

  float* ws   = (float*)d_ws;
  float* rA   = ws + OFF_RA;
  float* vA   = ws + OFF_VA;
  float* rB   = ws + OFF_RB;
  float* vB   = ws + OFF_VB;
  float* aacc = ws + OFF_AACC;
  float* mass = ws + OFF_MASS;
  float* sqw  = ws + OFF_SQ;
  float* wsum = ws + OFF_WSUM;
  float* feat = ws + OFF_FEAT;
  float* pdac = ws + OFF_PD;
  float* out  = (float*)d_out;

  init_kernel<<<MP, 128, 0, stream>>>(x, ap, am, iv, rA, vA, mass, pdac);
  for (int it = 0; it < ITERS; ++it) {
    prep_kernel<<<MP, 128, 0, stream>>>(rA, sqw, aacc, wsum);
    grav_kernel<<<dim3(MP / 64, JSPLIT), 128, 0, stream>>>(rA, sqw, mass, aacc, wsum);
    update_kernel<<<(MM * DD + 255) / 256, 256, 0, stream>>>(rA, vA, aacc, wsum, rB, vB);
    point_kernel<<<MM, 32, 0, stream>>>(rB, vB, rA, vA, bw1, bb1, bw2, bb2, bw3, bb3,
                                        pw1, pb1, pw2, kw1, kb1, kw2, (it % 5) == 0 ? 1 : 0);
  }
  stats1_kernel<<<BB, 128, 0, stream>>>(rA, am, feat);
  pdist_kernel<<<NN + KK, 128, 0, stream>>>(rA, pdac);
  final_kernel<<<1, 512, 0, stream>>>(pdac, feat, fw, fb, out);
}